// DepthwiseSeparableConv3d_14955076125374
// MI455X (gfx1250) — compile-verified
//
#include <hip/hip_runtime.h>

typedef __attribute__((ext_vector_type(2))) float v2f;
typedef __attribute__((ext_vector_type(4))) float v4f;
typedef __attribute__((ext_vector_type(8))) float v8f;

#define BN_EPS 1e-3f

// Problem dims (fixed by the reference)
constexpr int Bn = 2, Dd = 48, Hh = 48, Ww = 48, Cc = 64, Ff = 128;
constexpr int NV = Bn * Dd * Hh * Ww;          // 221184 voxels
constexpr int TILES = NV / 16;                 // 13824 wave-tiles (16 voxels along W)
constexpr int WAVES_PER_BLOCK = 4;             // 128 threads, wave32
constexpr int FRAG_STRIDE = 36;                // dwords per B fragment (32 data + 4 pad)
                                               // layout [kt][half][nt]: half gap = 8*36 = 288 ≡ 32 (mod 64)
                                               // -> the two 16-lane halves hit disjoint LDS bank sets
constexpr int YLDS_STRIDE = 68;                // padded row stride (floats) for per-wave y/z buffer
                                               // A-loads: 16 rows * 68 -> distinct banks; 8*68 ≡ 32 (mod 64)

__global__ __launch_bounds__(128) void dsconv3d_fused(
    const float* __restrict__ x,     // (B,D,H,W,C)
    const float* __restrict__ dwk,   // (3,3,3,1,C)
    const float* __restrict__ dwb,   // (C)
    const float* __restrict__ g1, const float* __restrict__ be1,
    const float* __restrict__ mu1, const float* __restrict__ va1,
    const float* __restrict__ pwk,   // (C,F)
    const float* __restrict__ pwb,   // (F)
    const float* __restrict__ g2, const float* __restrict__ be2,
    const float* __restrict__ mu2, const float* __restrict__ va2,
    float* __restrict__ out)         // (B,D,H,W,F)
{
    // B matrix (BN2-folded) pre-packed as WMMA fragments: one contiguous float2 per lane.
    // fragment (kt, half, nt): lane n16, elem v  holds  W[4*kt+2*half+v][nt*16+n16] * inv2[f]
    __shared__ float wfrag[16 * 2 * 8 * FRAG_STRIDE];               // 36864 B
    __shared__ float ybuf[WAVES_PER_BLOCK * 16 * YLDS_STRIDE];      // 17408 B (y, then z-transpose)

    const int tid = threadIdx.x;

    // ---- stage BN2-scaled pointwise weights into fragment-major LDS ----
    for (int i = tid; i < Cc * Ff; i += 128) {
        const int v    = i & 1;
        const int n16  = (i >> 1) & 15;
        const int nt   = (i >> 5) & 7;
        const int half = (i >> 8) & 1;
        const int kt   = i >> 9;
        const int k = 4 * kt + 2 * half + v;
        const int f = nt * 16 + n16;
        const float inv2 = g2[f] * rsqrtf(va2[f] + BN_EPS);
        wfrag[((kt * 2 + half) * 8 + nt) * FRAG_STRIDE + 2 * n16 + v] =
            pwk[k * Ff + f] * inv2;
    }
    __syncthreads();

    const int lane = tid & 31;
    const int wave = tid >> 5;
    const int tile = blockIdx.x * WAVES_PER_BLOCK + wave;

    // flattened voxel = tile*16 + m ;  decompose (b,d,h,w0)
    int t = tile;
    const int w0 = (t % 3) * 16;  t /= 3;
    const int h  = t % Hh;        t /= Hh;
    const int d  = t % Dd;        t /= Dd;
    const int b  = t;

    float* yw = &ybuf[wave * 16 * YLDS_STRIDE];

    // ================= depthwise conv + BN1 + ReLU -> LDS =================
    // lane owns channels c0 = lane, c1 = lane+32 (coalesced 128B x-loads)
    const int c0 = lane, c1 = lane + 32;

    const float inv1a = g1[c0] * rsqrtf(va1[c0] + BN_EPS);
    const float add1a = be1[c0] - mu1[c0] * inv1a;
    const float inv1b = g1[c1] * rsqrtf(va1[c1] + BN_EPS);
    const float add1b = be1[c1] - mu1[c1] * inv1b;

    float acc0[16], acc1[16];
    {
        const float bia0 = dwb[c0], bia1 = dwb[c1];
        #pragma unroll
        for (int m = 0; m < 16; ++m) { acc0[m] = bia0; acc1[m] = bia1; }
    }

    const bool lo_ok = (w0 != 0);    // w0-1  in range?
    const bool hi_ok = (w0 != 32);   // w0+16 in range?

    for (int kd = -1; kd <= 1; ++kd) {
        const int zd = d + kd;
        if ((unsigned)zd >= (unsigned)Dd) continue;          // wave-uniform
        for (int kh = -1; kh <= 1; ++kh) {
            const int zh = h + kh;
            if ((unsigned)zh >= (unsigned)Hh) continue;      // wave-uniform
            // cache 18-wide x row in registers (zero-filled at W borders; SAME = zero pad)
            const float* xrow =
                x + ((((size_t)b * Dd + zd) * Hh + zh) * Ww + (w0 - 1)) * Cc;
            float r0[18], r1[18];
            #pragma unroll
            for (int j = 0; j < 18; ++j) {
                const bool ok = (j > 0 || lo_ok) && (j < 17 || hi_ok);
                r0[j] = ok ? xrow[j * Cc + c0] : 0.0f;
                r1[j] = ok ? xrow[j * Cc + c1] : 0.0f;
            }
            const int tb = ((kd + 1) * 3 + (kh + 1)) * 3;
            float t0[3], t1[3];
            #pragma unroll
            for (int kw = 0; kw < 3; ++kw) {
                t0[kw] = dwk[(tb + kw) * Cc + c0];
                t1[kw] = dwk[(tb + kw) * Cc + c1];
            }
            #pragma unroll
            for (int m = 0; m < 16; ++m) {
                #pragma unroll
                for (int kw = 0; kw < 3; ++kw) {
                    acc0[m] = fmaf(r0[m + kw], t0[kw], acc0[m]);
                    acc1[m] = fmaf(r1[m + kw], t1[kw], acc1[m]);
                }
            }
        }
    }

    #pragma unroll
    for (int m = 0; m < 16; ++m) {
        yw[m * YLDS_STRIDE + c0] = fmaxf(fmaf(acc0[m], inv1a, add1a), 0.0f);
        yw[m * YLDS_STRIDE + c1] = fmaxf(fmaf(acc1[m], inv1b, add1b), 0.0f);
    }
    // same-wave LDS RAW: compiler inserts s_wait_dscnt; no barrier needed.

    // ================= pointwise GEMM: [16 x 64] @ [64 x 128] via WMMA f32 =================
    const int half = lane >> 4;   // 0: lanes 0-15, 1: lanes 16-31
    const int lml  = lane & 15;

    v8f acc[8];
    #pragma unroll
    for (int nt = 0; nt < 8; ++nt) {
        const int f = nt * 16 + lml;
        const float inv2 = g2[f] * rsqrtf(va2[f] + BN_EPS);
        const float bias2 = fmaf(pwb[f] - mu2[f], inv2, be2[f]);
        #pragma unroll
        for (int r = 0; r < 8; ++r) acc[nt][r] = bias2;
    }

    #pragma unroll 4
    for (int kt = 0; kt < 16; ++kt) {
        // A 16x4 fragment: lane holds M=lml, K = 4*kt + 2*half + {0,1}   (8B-aligned b64)
        const v2f afrag = *(const v2f*)&yw[lml * YLDS_STRIDE + 4 * kt + 2 * half];
        #pragma unroll
        for (int nt = 0; nt < 8; ++nt) {
            // one contiguous b64 per lane, even-aligned, bank-conflict free
            const v2f bfrag =
                *(const v2f*)&wfrag[((kt * 2 + half) * 8 + nt) * FRAG_STRIDE + 2 * lml];
            acc[nt] = __builtin_amdgcn_wmma_f32_16x16x4_f32(
                false, afrag, false, bfrag, (short)0, acc[nt], false, false);
        }
    }

    // ================= ReLU + transpose through LDS + b128 stores =================
    // y region is dead now; reuse it as a 16 x 64 z-transpose buffer, two column passes.
    const size_t obase = (size_t)tile * 16 * Ff;
    #pragma unroll
    for (int p = 0; p < 2; ++p) {
        #pragma unroll
        for (int nt4 = 0; nt4 < 4; ++nt4) {
            const int nt = p * 4 + nt4;
            #pragma unroll
            for (int r = 0; r < 8; ++r) {
                // halves write rows r and r+8: 8*68 ≡ 32 (mod 64) -> disjoint banks
                yw[(r + 8 * half) * YLDS_STRIDE + nt4 * 16 + lml] =
                    fmaxf(acc[nt][r], 0.0f);
            }
        }
        // each iteration: wave stores 512B as 2 rows x 256B half-row segments
        #pragma unroll
        for (int it = 0; it < 8; ++it) {
            const int m = it * 2 + half;          // lanes 0-15 -> row 2it, 16-31 -> 2it+1
            const v4f vals = *(const v4f*)&yw[m * YLDS_STRIDE + lml * 4];
            *(v4f*)&out[obase + (size_t)m * Ff + p * 64 + lml * 4] = vals;
        }
    }
}

extern "C" void kernel_launch(void* const* d_in, const int* in_sizes, int n_in,
                              void* d_out, int out_size, void* d_ws, size_t ws_size,
                              hipStream_t stream) {
    const float* x    = (const float*)d_in[0];
    const float* dwk  = (const float*)d_in[1];
    const float* dwb  = (const float*)d_in[2];
    const float* g1   = (const float*)d_in[3];
    const float* be1  = (const float*)d_in[4];
    const float* mu1  = (const float*)d_in[5];
    const float* va1  = (const float*)d_in[6];
    const float* pwk  = (const float*)d_in[7];
    const float* pwb  = (const float*)d_in[8];
    const float* g2   = (const float*)d_in[9];
    const float* be2  = (const float*)d_in[10];
    const float* mu2  = (const float*)d_in[11];
    const float* va2  = (const float*)d_in[12];
    float* out = (float*)d_out;

    const int blocks = TILES / WAVES_PER_BLOCK;   // 13824 / 4 = 3456, exact
    dsconv3d_fused<<<blocks, 128, 0, stream>>>(
        x, dwk, dwb, g1, be1, mu1, va1, pwk, pwb, g2, be2, mu2, va2, out);
}